// SudokuLoss_85074712199409
// MI455X (gfx1250) — compile-verified
//
#include <hip/hip_runtime.h>
#include <stdint.h>

// ---------------- problem constants (match reference) ----------------
#define NB        65536          // batches
#define CELLS     81             // 9x9 cells per batch
#define LOGB      729            // 81 cells * 9 classes
#define TILE_B    16             // batches staged per LDS tile (keeps tile base 16B aligned: 16*729*4 = 46656)
#define BLOCK     256            // 8 waves (wave32)
#define NWG       1024
#define NTILES    (NB / TILE_B)              // 4096 -> 4 tiles per WG
#define V4_TILE   (TILE_B * LOGB / 4)        // 2916 b128 transfers per tile
#define CB_TILE   (TILE_B * 162)             // 81 col + 81 box accumulators per batch

// ============================================================================
// Kernel 1: per-workgroup partial sums
//   partials[wg*8 + {0..4}] = {focal_sum, mask_count, entropy_sum,
//                              uniqueness_sum, rowcolbox_sq_sum}
// ============================================================================
__global__ __launch_bounds__(BLOCK) void sudoku_partial(
    const float* __restrict__ logits,
    const int*   __restrict__ targets,
    const int*   __restrict__ puzzles,
    float*       __restrict__ partials)
{
  __shared__ __align__(16) float sh_logits[TILE_B * LOGB];   // 46,656 B
  __shared__ __align__(16) float sh_cb[CB_TILE];             // 10,368 B (also reused for final reduction)

  const int  tid    = threadIdx.x;
  const int  bl     = tid / 9;          // batch within tile (valid when active)
  const int  row    = tid - bl * 9;     // sudoku row owned by this thread
  const bool active = (tid < TILE_B * 9);   // 144 compute threads; all 256 stage/zero/square

  // low 32 bits of a generic LDS pointer == wave-relative LDS byte offset (aperture layout)
  const uint32_t lds_log_base = (uint32_t)(uint64_t)(uintptr_t)&sh_logits[0];

  float acc_focal = 0.f, acc_mask = 0.f, acc_ent = 0.f, acc_uniq = 0.f, acc_sq = 0.f;

  for (int tile = blockIdx.x; tile < NTILES; tile += NWG) {
    const size_t tb    = (size_t)tile * TILE_B;
    const float* gbase = logits + tb * LOGB;

    // ---- async stage: global -> LDS, 16 B per lane per op (CDNA5 async path) ----
    for (int k = tid; k < V4_TILE; k += BLOCK) {
      const uint64_t ga = (uint64_t)(const void*)(gbase + (size_t)k * 4);
      const uint32_t lo = lds_log_base + (uint32_t)k * 16u;
      asm volatile("global_load_async_to_lds_b128 %0, %1, off"
                   :: "v"(lo), "v"(ga) : "memory");
    }
    // zero the col/box accumulators while the async copies are in flight
    for (int k = tid; k < CB_TILE; k += BLOCK) sh_cb[k] = 0.f;

    asm volatile("s_wait_asynccnt 0" ::: "memory");
    __syncthreads();

    // ---- compute phase: one thread per (batch, row) ----
    if (active) {
      const size_t b    = tb + (size_t)bl;
      const float* xr   = &sh_logits[bl * LOGB + row * 81];
      const int*   tg   = targets + b * CELLS + row * 9;
      const int*   pz   = puzzles + b * CELLS + row * 9;
      float*       colp = &sh_cb[bl * 162];     // [9 cols][9 classes]
      float*       boxp = colp + 81;            // [9 boxes][9 classes]

      float rowsum[9];
      #pragma unroll
      for (int c = 0; c < 9; ++c) rowsum[c] = 0.f;
      float bpart[27];                           // 3 box-columns x 9 classes (this row's band)
      #pragma unroll
      for (int k = 0; k < 27; ++k) bpart[k] = 0.f;

      for (int s = 0; s < 9; ++s) {
        int j = row + s; if (j >= 9) j -= 9;    // stagger: 9 row-threads hit distinct columns each step
        const float* x = xr + j * 9;
        float xv[9];
        #pragma unroll
        for (int c = 0; c < 9; ++c) xv[c] = x[c];

        const int   pzv = pz[j];
        int t = tg[j] - 1; t = t < 0 ? 0 : (t > 8 ? 8 : t);   // clip(targets-1, 0, 8)
        const float m = (pzv == 0) ? 1.f : 0.f;

        // numerically-stable softmax over 9 classes
        float mx = xv[0];
        #pragma unroll
        for (int c = 1; c < 9; ++c) mx = fmaxf(mx, xv[c]);
        float e[9]; float Z = 0.f;
        #pragma unroll
        for (int c = 0; c < 9; ++c) { e[c] = __expf(xv[c] - mx); Z += e[c]; }
        const float rZ   = 1.f / Z;
        const float logZ = __logf(Z);

        float ent = 0.f, p1 = -1.f, p2 = -1.f, lpt = 0.f, pt = 0.f;
        const int bc = j / 3;
        #pragma unroll
        for (int c = 0; c < 9; ++c) {
          const float pc  = e[c] * rZ;                 // prob
          const float lpc = (xv[c] - mx) - logZ;       // log prob
          ent = fmaf(-pc, lpc, ent);                   // entropy accum
          const bool ist = (c == t);                   // select target w/o dynamic reg index
          lpt = ist ? lpc : lpt;
          pt  = ist ? pc  : pt;
          const float mn = fminf(pc, p1);              // top-2 tracking
          p1 = fmaxf(pc, p1);
          p2 = fmaxf(p2, mn);
          const float mp = pc * m;                     // masked prob
          rowsum[c]        += mp;                      // row sums in regs
          atomicAdd(&colp[j * 9 + c], mp);             // ds_add_f32 (collision-free per step)
          bpart[bc * 9 + c] += mp;                     // box partials in regs
        }
        const float omp = 1.f - pt;
        acc_focal += m * (omp * omp * (-lpt));         // (1-pt)^2 * CE, masked
        acc_mask  += m;
        acc_ent   += m * ent;
        const float u = 1.f - (p1 - p2);               // relu(1 - gap)
        acc_uniq  += (u > 0.f) ? u : 0.f;
      }

      // row constraint: (rowsum - 1)^2
      #pragma unroll
      for (int c = 0; c < 9; ++c) { const float d = rowsum[c] - 1.f; acc_sq = fmaf(d, d, acc_sq); }
      // fold this row's band partials into the shared box accumulators
      const int bi = row / 3;
      #pragma unroll
      for (int bc2 = 0; bc2 < 3; ++bc2)
        #pragma unroll
        for (int c = 0; c < 9; ++c)
          atomicAdd(&boxp[(bi * 3 + bc2) * 9 + c], bpart[bc2 * 9 + c]);
    }
    __syncthreads();

    // ---- col + box constraint squared errors (all 256 threads) ----
    for (int k = tid; k < CB_TILE; k += BLOCK) {
      const float d = sh_cb[k] - 1.f;
      acc_sq = fmaf(d, d, acc_sq);
    }
    __syncthreads();   // protect sh_cb re-zero AND sh_logits re-stage of the next tile
  }

  // ---- block tree-reduction of the 5 partial scalars (reuse sh_cb) ----
  float vals[5] = {acc_focal, acc_mask, acc_ent, acc_uniq, acc_sq};
  #pragma unroll
  for (int v = 0; v < 5; ++v) {
    sh_cb[tid] = vals[v];
    __syncthreads();
    for (int off = BLOCK / 2; off > 0; off >>= 1) {
      if (tid < off) sh_cb[tid] += sh_cb[tid + off];
      __syncthreads();
    }
    if (tid == 0) partials[(size_t)blockIdx.x * 8 + v] = sh_cb[0];
    __syncthreads();
  }
}

// ============================================================================
// Kernel 2: deterministic fixed-order combine of the 1024 WG partials
// ============================================================================
__global__ __launch_bounds__(BLOCK) void sudoku_finalize(
    const float* __restrict__ partials, float* __restrict__ out)
{
  __shared__ float red[BLOCK];
  const int tid = threadIdx.x;
  float s[5] = {0.f, 0.f, 0.f, 0.f, 0.f};
  for (int w = tid; w < NWG; w += BLOCK) {
    #pragma unroll
    for (int v = 0; v < 5; ++v) s[v] += partials[(size_t)w * 8 + v];
  }
  float tot[5];
  for (int v = 0; v < 5; ++v) {
    red[tid] = s[v];
    __syncthreads();
    for (int off = BLOCK / 2; off > 0; off >>= 1) {
      if (tid < off) red[tid] += red[tid + off];
      __syncthreads();
    }
    tot[v] = red[0];
    __syncthreads();
  }
  if (tid == 0) {
    const float F = tot[0], M = tot[1], E = tot[2], U = tot[3], SQ = tot[4];
    const float inv_m   = 1.f / (M + 1e-8f);
    const float ncells  = (float)NB * 81.f;                 // exact in f32 (< 2^24)
    const float ce_loss = F * inv_m;
    const float rcb     = SQ / ncells;                      // row+col+box share normalizer & weight
    const float ent_l   = 0.1f * E * inv_m;
    const float uniq_l  = 0.1f * (U / ncells);
    const float constraint = (rcb + ent_l + uniq_l) * 0.2f;
    out[0] = ce_loss + 0.5f * constraint;
  }
}

// ============================================================================
extern "C" void kernel_launch(void* const* d_in, const int* in_sizes, int n_in,
                              void* d_out, int out_size, void* d_ws, size_t ws_size,
                              hipStream_t stream) {
  (void)in_sizes; (void)n_in; (void)out_size; (void)ws_size;
  const float* logits  = (const float*)d_in[0];
  const int*   targets = (const int*)d_in[1];
  const int*   puzzles = (const int*)d_in[2];
  float*       part    = (float*)d_ws;          // NWG * 8 floats = 32 KB scratch

  sudoku_partial<<<NWG, BLOCK, 0, stream>>>(logits, targets, puzzles, part);
  sudoku_finalize<<<1, BLOCK, 0, stream>>>(part, (float*)d_out);
}